// EfficientKAN_13958643712426
// MI455X (gfx1250) — compile-verified
//
#include <hip/hip_runtime.h>

typedef __attribute__((ext_vector_type(2))) float v2f;
typedef __attribute__((ext_vector_type(8))) float v8f;

#define DIM     128
#define NB      6
#define OUTW    (DIM * NB + DIM)   // 896 output columns
#define ROWS    16                 // batch rows per block
#define XXSTR   130                // padded LDS stride (keeps 8B align, kills bank conflicts)

__global__ __launch_bounds__(256)
void kan_fused_kernel(const float* __restrict__ x,
                      const float* __restrict__ W,
                      float* __restrict__ out)
{
    __shared__ float lds_xx[ROWS * XXSTR];   // x^2 tile (A operand), padded
    __shared__ float lds_S[DIM * DIM];       // sigmoid(W), row-major (B operand)

    const int t    = threadIdx.x;
    const int row0 = blockIdx.x * ROWS;

    // ---- Stage S = sigmoid(W) into LDS (64KB), coalesced; W stays L2-resident ----
    #pragma unroll
    for (int i = 0; i < (DIM * DIM) / 256; ++i) {
        int idx = t + i * 256;
        float w = W[idx];
        lds_S[idx] = 1.0f / (1.0f + __expf(-w));
    }

    const float h = 2.0f / 9.0f;  // uniform knot spacing, knots = -1 + j*h, j=0..9

    // ---- Load x tile, deposit x^2 to LDS, compute degree-3 B-spline basis, store ----
    #pragma unroll
    for (int i = 0; i < (ROWS * DIM) / 256; ++i) {   // 8 iterations
        int idx = t + i * 256;
        int r = idx >> 7;        // 0..15
        int c = idx & 127;       // 0..127
        float xv = x[(size_t)(row0 + r) * DIM + c];
        lds_xx[r * XXSTR + c] = xv * xv;

        // Cox-de Boor on 10 uniform knots; all denominators = d*h > 0.
        float xc = fminf(fmaxf(xv, -1.0f), 1.0f - 1e-6f);
        float Bv[9];
        #pragma unroll
        for (int j = 0; j < 9; ++j) {
            float kl = -1.0f + j * h;
            float kr = -1.0f + (j + 1) * h;
            Bv[j] = (xc >= kl && xc < kr) ? 1.0f : 0.0f;
        }
        #pragma unroll
        for (int d = 1; d <= 3; ++d) {
            float inv = 1.0f / (d * h);
            #pragma unroll
            for (int j = 0; j <= 8 - d; ++j) {   // in-place: Bv[j+1] still degree d-1
                float kj   = -1.0f + j * h;
                float kjd1 = -1.0f + (j + d + 1) * h;
                Bv[j] = (xc - kj) * inv * Bv[j] + (kjd1 - xc) * inv * Bv[j + 1];
            }
        }
        // out[b, c*6 .. c*6+5]; 8B-aligned -> three b64 stores, dense across the wave
        float2* o = reinterpret_cast<float2*>(out + (size_t)(row0 + r) * OUTW + c * NB);
        o[0] = make_float2(Bv[0], Bv[1]);
        o[1] = make_float2(Bv[2], Bv[3]);
        o[2] = make_float2(Bv[4], Bv[5]);
    }

    __syncthreads();

    // ---- interaction[16 x 128] = xx[16x128] @ S[128x128] via V_WMMA_F32_16X16X4_F32 ----
    // Wave w owns N-tile [16w, 16w+16). EXEC is all 1s here (required by WMMA).
    const int lane  = t & 31;
    const int wave  = t >> 5;
    const int half  = lane >> 4;     // 0: K pair {k,k+1}; 1: K pair {k+2,k+3}
    const int m     = lane & 15;     // A: M index; B: N offset
    const int n     = wave * 16 + m;
    const int koff  = half * 2;

    v8f acc = {};
    #pragma unroll 4
    for (int k = 0; k < DIM; k += 4) {
        int kk = k + koff;
        v2f a, b;
        // A 16x4: lane<16 -> M=lane, VGPR0=K, VGPR1=K+1 ; lane>=16 -> K+2, K+3
        a.x = lds_xx[m * XXSTR + kk];
        a.y = lds_xx[m * XXSTR + kk + 1];
        // B 4x16: lane<16 -> N=lane, K / K+1 ; lane>=16 -> same N, K+2 / K+3
        b.x = lds_S[kk * DIM + n];
        b.y = lds_S[(kk + 1) * DIM + n];
        acc = __builtin_amdgcn_wmma_f32_16x16x4_f32(false, a, false, b,
                                                    (short)0, acc, false, false);
    }

    // C/D 16x16 f32: VGPR v -> M = v + 8*half, N = n
    float* obase = out + (size_t)row0 * OUTW + DIM * NB + n;
    #pragma unroll
    for (int v = 0; v < 8; ++v) {
        int mm = v + 8 * half;
        obase[(size_t)mm * OUTW] = acc[v];
    }
}

extern "C" void kernel_launch(void* const* d_in, const int* in_sizes, int n_in,
                              void* d_out, int out_size, void* d_ws, size_t ws_size,
                              hipStream_t stream) {
    const float* x = (const float*)d_in[0];            // [B, 128] f32
    const float* W = (const float*)d_in[1];            // [128, 128] f32
    float* out = (float*)d_out;                        // [B, 896] f32

    const int batch = in_sizes[0] / DIM;               // 65536
    dim3 grid(batch / ROWS), block(256);
    kan_fused_kernel<<<grid, block, 0, stream>>>(x, W, out);
}